// NeuralMemory_9586367005219
// MI455X (gfx1250) — compile-verified
//
#include <hip/hip_runtime.h>
#include <hip/hip_bf16.h>

typedef float v2f __attribute__((ext_vector_type(2)));
typedef float v4f __attribute__((ext_vector_type(4)));
typedef float v8f __attribute__((ext_vector_type(8)));

// Problem constants
static constexpr int   D        = 128;
static constexpr int   T        = 2048;          // B*N tokens
static constexpr size_t SZ      = (size_t)T * D; // one per-token vector array
static constexpr size_t CUM_ELEMS = 4ull * 2 * 1024 * 128 * 128; // 134217728
static constexpr size_t NM_OFF    = CUM_ELEMS;                    // next_mem offset
static constexpr size_t LOSS_OFF  = CUM_ELEMS + 4ull * 2 * 128 * 128; // losses offset

enum { EP_NONE = 0, EP_SILU = 1, EP_DSILU = 2 };

// ---------------------------------------------------------------------------
// Phase 1: fp32 WMMA GEMM, one wave per 16x128 strip (K = 128 fixed).
//   TRANSB=true : O = A(MxK) @ Bw(NxK)^T   (forward, weight rows)
//   TRANSB=false: O = A(MxK) @ Bw(KxN)     (backward delta propagation)
// Epilogues: NONE -> store h; SILU -> store h and silu(h);
//            DSILU -> store h * silu'(Hsrc)   (elementwise)
//
// A-fragments for the whole strip are hoisted into registers (32 x v2f) so
// the k-loop issues only B loads; B loads are batched per k-step ahead of the
// 8 WMMAs so the scheduler can overlap loads with v_wmma instead of
// wait(0)-per-load serialization.
// ---------------------------------------------------------------------------
template <bool TRANSB, int EPI>
__global__ __launch_bounds__(32) void gemm_wmma(
    const float* __restrict__ A, const float* __restrict__ Bw,
    float* __restrict__ O0, float* __restrict__ O1,
    const float* __restrict__ Hsrc)
{
    constexpr int K = 128, N = 128;
    const int m0   = blockIdx.x * 16;
    const int lane = threadIdx.x;
    const int half = lane >> 4;     // 0: lanes 0-15, 1: lanes 16-31
    const int lm   = lane & 15;

    // Hoist all A fragments: af[kk] covers K = 4*kk .. 4*kk+3
    // layout: VGPR0 = K k0+2*half, VGPR1 = K k0+2*half+1
    v2f af[K / 4];
    {
        const float* arow = A + (size_t)(m0 + lm) * K + 2 * half;
#pragma unroll
        for (int kk = 0; kk < K / 4; ++kk)
            af[kk] = *(const v2f*)(arow + 4 * kk);
    }

    v8f acc[8];
#pragma unroll
    for (int i = 0; i < 8; ++i) acc[i] = (v8f)0.f;

    for (int kk = 0; kk < K / 4; ++kk) {
        const int ka = 4 * kk + 2 * half;
        // Batch all 8 B fragments for this k-step, then run the 8 WMMAs.
        v2f bf[8];
#pragma unroll
        for (int nt = 0; nt < 8; ++nt) {
            const int n = nt * 16 + lm;
            if (TRANSB) {
                bf[nt] = *(const v2f*)(Bw + (size_t)n * K + ka);
            } else {
                bf[nt].x = Bw[(size_t)ka * N + n];
                bf[nt].y = Bw[(size_t)(ka + 1) * N + n];
            }
        }
#pragma unroll
        for (int nt = 0; nt < 8; ++nt) {
            acc[nt] = __builtin_amdgcn_wmma_f32_16x16x4_f32(
                false, af[kk], false, bf[nt], (short)0, acc[nt], false, false);
        }
    }

    // Epilogue: C/D layout -> VGPR v holds rows v (lanes 0-15) and v+8 (16-31)
#pragma unroll
    for (int nt = 0; nt < 8; ++nt) {
#pragma unroll
        for (int v = 0; v < 8; ++v) {
            const int row = m0 + v + 8 * half;
            const int col = nt * 16 + lm;
            const size_t off = (size_t)row * N + col;
            const float h = acc[nt][v];
            if (EPI == EP_NONE) {
                O0[off] = h;
            } else if (EPI == EP_SILU) {
                const float sg = 1.f / (1.f + __expf(-h));
                O0[off] = h;
                O1[off] = h * sg;
            } else { // EP_DSILU
                const float hh = Hsrc[off];
                const float sg = 1.f / (1.f + __expf(-hh));
                const float ds = sg * (1.f + hh * (1.f - sg));
                O0[off] = h * ds;
            }
        }
    }
}

// ---------------------------------------------------------------------------
// d3 = 2*(pred - v)/D ; losses[t] = mean((pred-v)^2)
// ---------------------------------------------------------------------------
__global__ __launch_bounds__(128) void loss_d3_kernel(
    const float* __restrict__ pred, const float* __restrict__ vals,
    float* __restrict__ d3, float* __restrict__ losses)
{
    __shared__ float red[128];
    const int t = blockIdx.x;
    const int i = threadIdx.x;
    const size_t off = (size_t)t * D + i;
    const float e = pred[off] - vals[off];
    d3[off] = e * (2.f / (float)D);
    red[i] = e * e;
    __syncthreads();
    for (int s = 64; s > 0; s >>= 1) {
        if (i < s) red[i] += red[i + s];
        __syncthreads();
    }
    if (i == 0) losses[t] = red[0] * (1.f / (float)D);
}

// delta / act selection per matrix index m (offsets in units of SZ)
__device__ __forceinline__ const float* delta_ptr(const float* ws, int m) {
    return ws + (size_t)(12 - m) * SZ;            // d0=12,d1=11,d2=10,d3=9
}
__device__ __forceinline__ const float* act_ptr(const float* ws, int m) {
    return ws + (size_t)(m == 0 ? 0 : 2 * m + 1) * SZ; // keys=0,a0=3,a1=5,a2=7
}

// ---------------------------------------------------------------------------
// Phase 2a: per-chunk (128-token) outer-product sums -> chunks scratch
// grid: 1024 = m(4) x b(2) x chunk(8) x rowtile(16 of 8 rows). block: 256.
// ---------------------------------------------------------------------------
__global__ __launch_bounds__(256) void p2a_kernel(
    const float* __restrict__ ws, float* __restrict__ chunks)
{
    const int bx = blockIdx.x;
    const int rt = bx & 15, ch = (bx >> 4) & 7, bb = (bx >> 7) & 1, m = bx >> 8;
    const float* delta = delta_ptr(ws, m);
    const float* act   = act_ptr(ws, m);
    const int r = threadIdx.x >> 5;
    const int c = (threadIdx.x & 31) * 4;
    const int row = rt * 8 + r;

    v4f acc = (v4f)0.f;
    const int tbase = bb * 1024 + ch * 128;
    for (int t = 0; t < 128; ++t) {
        const size_t g = (size_t)(tbase + t) * D;
        const float dv = delta[g + row];
        const v4f a4 = *(const v4f*)(act + g + c);
        acc += dv * a4;
    }
    const size_t off = ((size_t)((m * 2 + bb) * 8 + ch)) * 16384 + (size_t)row * 128 + c;
    *(v4f*)(chunks + off) = acc;
}

// ---------------------------------------------------------------------------
// Phase 2b: seeded local cumsum, streams 512 MB of NT stores; ch==7 blocks
// also emit next_mem = W + total.
// ---------------------------------------------------------------------------
__global__ __launch_bounds__(256) void p2b_kernel(
    const float* __restrict__ ws, const float* __restrict__ chunks,
    const float* __restrict__ W0, const float* __restrict__ W1,
    const float* __restrict__ W2, const float* __restrict__ W3,
    float* __restrict__ out)
{
    const int bx = blockIdx.x;
    const int rt = bx & 15, ch = (bx >> 4) & 7, bb = (bx >> 7) & 1, m = bx >> 8;
    const float* delta = delta_ptr(ws, m);
    const float* act   = act_ptr(ws, m);
    const int r = threadIdx.x >> 5;
    const int c = (threadIdx.x & 31) * 4;
    const int row = rt * 8 + r;
    const int mb = m * 2 + bb;

    v4f acc = (v4f)0.f;
    for (int pc = 0; pc < ch; ++pc) {  // exclusive prefix of earlier chunks
        const float* cp = chunks + ((size_t)(mb * 8 + pc)) * 16384 + (size_t)row * 128 + c;
        acc += *(const v4f*)cp;
    }

    for (int t = 0; t < 128; ++t) {
        const int n = ch * 128 + t;
        const size_t g = (size_t)(bb * 1024 + n) * D;
        const float dv = delta[g + row];
        const v4f a4 = *(const v4f*)(act + g + c);
        acc += dv * a4;
        float* dst = out + ((size_t)(mb * 1024 + n)) * 16384 + (size_t)row * 128 + c;
        __builtin_nontemporal_store(acc, (v4f*)dst);  // streaming store, skip cache
    }

    if (ch == 7) {  // final running sum == total -> next_mem = W + total
        const float* W = (m == 0) ? W0 : (m == 1) ? W1 : (m == 2) ? W2 : W3;
        const size_t wo = (size_t)row * 128 + c;
        v4f o = *(const v4f*)(W + wo) + acc;
        *(v4f*)(out + NM_OFF + (size_t)mb * 16384 + wo) = o;
    }
}

// ---------------------------------------------------------------------------
extern "C" void kernel_launch(void* const* d_in, const int* in_sizes, int n_in,
                              void* d_out, int out_size, void* d_ws, size_t ws_size,
                              hipStream_t stream)
{
    const float* seq = (const float*)d_in[0];   // (2,1024,128)
    const float* Wkv = (const float*)d_in[1];   // (256,128)
    const float* W0  = (const float*)d_in[2];
    const float* W1  = (const float*)d_in[3];
    const float* W2  = (const float*)d_in[4];
    const float* W3  = (const float*)d_in[5];
    float* out = (float*)d_out;
    float* ws  = (float*)d_ws;

    // Workspace layout (floats): 13 per-token vector arrays + chunk sums
    float* keys   = ws + 0 * SZ;
    float* values = ws + 1 * SZ;
    float* h0     = ws + 2 * SZ;
    float* a0     = ws + 3 * SZ;
    float* h1     = ws + 4 * SZ;
    float* a1     = ws + 5 * SZ;
    float* h2     = ws + 6 * SZ;
    float* a2     = ws + 7 * SZ;
    float* pred   = ws + 8 * SZ;
    float* d3     = ws + 9 * SZ;
    float* d2     = ws + 10 * SZ;
    float* d1     = ws + 11 * SZ;
    float* d0     = ws + 12 * SZ;
    float* chunks = ws + 13 * SZ;   // 4*2*8*128*128 floats

    const dim3 gg(T / 16), gb(32);

    // Phase 1: forward
    gemm_wmma<true,  EP_NONE ><<<gg, gb, 0, stream>>>(seq,  Wkv,         keys,   nullptr, nullptr);
    gemm_wmma<true,  EP_NONE ><<<gg, gb, 0, stream>>>(seq,  Wkv + D * D, values, nullptr, nullptr);
    gemm_wmma<true,  EP_SILU ><<<gg, gb, 0, stream>>>(keys, W0,          h0,     a0,      nullptr);
    gemm_wmma<true,  EP_SILU ><<<gg, gb, 0, stream>>>(a0,   W1,          h1,     a1,      nullptr);
    gemm_wmma<true,  EP_SILU ><<<gg, gb, 0, stream>>>(a1,   W2,          h2,     a2,      nullptr);
    gemm_wmma<true,  EP_NONE ><<<gg, gb, 0, stream>>>(a2,   W3,          pred,   nullptr, nullptr);

    // Loss + output delta
    loss_d3_kernel<<<T, 128, 0, stream>>>(pred, values, d3, out + LOSS_OFF);

    // Phase 1: backward deltas (non-transposed weight GEMMs + silu')
    gemm_wmma<false, EP_DSILU><<<gg, gb, 0, stream>>>(d3, W3, d2, nullptr, h2);
    gemm_wmma<false, EP_DSILU><<<gg, gb, 0, stream>>>(d2, W2, d1, nullptr, h1);
    gemm_wmma<false, EP_DSILU><<<gg, gb, 0, stream>>>(d1, W1, d0, nullptr, h0);

    // Phase 2: chunked cumsum of rank-1 outer products (store-bandwidth bound)
    p2a_kernel<<<1024, 256, 0, stream>>>(ws, chunks);
    p2b_kernel<<<1024, 256, 0, stream>>>(ws, chunks, W0, W1, W2, W3, out);
}